// MVDR_Oracle_no_steering_vector_68135361183974
// MI455X (gfx1250) — compile-verified
//
#include <hip/hip_runtime.h>

// ---------------------------------------------------------------------------
// MVDR (oracle noise covariance) for MI455X / gfx1250, wave32.
//
// Shapes: mixture/noise [N=8, 2, C=8, T=500, F=513] f32, out [8,2,1,500,513].
//
// Memory-bound (~0.4 GB traffic vs ~2.3 GFLOP f32) => keep f32 precision
// (8x8 inversion downstream) and optimize line utilization.
//
// Covariance: S = [Yre;Yim]*[Yre;Yim]^T (16x16x500 f32 GEMM per (n,f)) via
// V_WMMA_F32_16X16X4_F32. With the CDNA5 f32 A(16x4)/B(4x16) layouts the
// symmetric Gram uses identical VGPRs for A and B. Each wave processes FOUR
// consecutive f: a lane loads a 16B float4 along f at its (row,t) slot and
// feeds 4 independent WMMA chains from the components — 16B requests, and a
// workgroup (8 waves = 32 f) covers full 128B cachelines.
// ---------------------------------------------------------------------------

typedef float v2f __attribute__((ext_vector_type(2)));
typedef float v4f __attribute__((ext_vector_type(4)));
typedef float v8f __attribute__((ext_vector_type(8)));

#define NB 8
#define CB 8
#define TB 500
#define FB 513
#define NF (NB * FB) /* 4104 independent (n,f) problems */
#define QN 129       /* f-chunks of 4 per batch: q<128 -> f0=4q, q==128 -> f0=509 */

// Workspace layout in floats:
#define SY_OFF 0                 /* 4104 * 256 f32  : S matrices of mixture */
#define SV_OFF (NF * 256)        /* 4104 * 256 f32  : S matrices of noise   */
#define H_OFF  (2 * NF * 256)    /* 4104 * 8 float2 : conj(h) weights       */

// align-4-safe 16B load (backend may emit b128; split loads also correct)
__device__ __forceinline__ v4f ld4(const float* p) {
  v4f v;
  __builtin_memcpy(&v, p, 16);
  return v;
}

// ---------------------------------------------------------------------------
// Kernel 1: covariance Grams via f32 WMMA. One wave per (n, 4-f chunk).
// 129 blocks * 8 waves = 1032 waves == NB*QN exactly (full EXEC for WMMA).
// ---------------------------------------------------------------------------
__global__ __launch_bounds__(256) void cov_wmma_kernel(
    const float* __restrict__ mix, const float* __restrict__ noi,
    float* __restrict__ ws) {
  const int lane   = threadIdx.x & 31;
  const int waveId = blockIdx.x * 8 + (threadIdx.x >> 5);
  const int n  = waveId / QN;
  const int q  = waveId - n * QN;
  const int f0 = (q == QN - 1) ? (FB - 4) : 4 * q;  // tail overlaps (benign)

  const int hh = lane >> 4;   // lane half: K offset {0,2}
  const int r  = lane & 15;   // row M (A) == col N (B) for this lane
  const int ri = r >> 3;      // 0 = real rows, 1 = imag rows
  const int c  = r & 7;       // mic channel

  const long base = (long)n * (2L * CB * TB * FB) + (long)ri * (CB * TB * FB) +
                    (long)c * ((long)TB * FB) + f0;
  const float* pm = mix + base;
  const float* pn = noi + base;

  v8f sy[4] = {};  // 4 f-problems, mixture
  v8f sv[4] = {};  // 4 f-problems, noise
  const int t0 = hh * 2;

  for (int k = 0; k < TB; k += 4) {
    const long o0 = (long)(k + t0) * FB;
    // 16B along f at (row, t) and (row, t+1), for mixture and noise
    const v4f m0 = ld4(pm + o0);
    const v4f m1 = ld4(pm + o0 + FB);
    const v4f n0 = ld4(pn + o0);
    const v4f n1 = ld4(pn + o0 + FB);
#pragma unroll
    for (int j = 0; j < 4; ++j) {
      v2f am, an;
      am.x = m0[j];
      am.y = m1[j];
      an.x = n0[j];
      an.y = n1[j];
      // Symmetric Gram: A and B operands are the same registers.
      sy[j] = __builtin_amdgcn_wmma_f32_16x16x4_f32(false, am, false, am,
                                                    (short)0, sy[j], false,
                                                    false);
      sv[j] = __builtin_amdgcn_wmma_f32_16x16x4_f32(false, an, false, an,
                                                    (short)0, sv[j], false,
                                                    false);
    }
  }

  const float invT = 1.0f / (float)TB;  // time average
#pragma unroll
  for (int j = 0; j < 4; ++j) {
    const long idx = (long)n * FB + (f0 + j);
    float* Sy = ws + SY_OFF + idx * 256;
    float* Sv = ws + SV_OFF + idx * 256;
#pragma unroll
    for (int v = 0; v < 8; ++v) {
      // C/D f32 layout: VGPR v -> M = v + 8*(lane>=16), N = lane & 15
      const int m = v + 8 * hh;
      Sy[m * 16 + r] = sy[j][v] * invT;
      Sv[m * 16 + r] = sv[j][v] * invT;
    }
  }
}

// ---------------------------------------------------------------------------
// Complex helpers for the tiny per-(n,f) 8x8 solve.
// ---------------------------------------------------------------------------
__device__ __forceinline__ float2 cmul(float2 a, float2 b) {
  return make_float2(a.x * b.x - a.y * b.y, a.x * b.y + a.y * b.x);
}
__device__ __forceinline__ float2 csub(float2 a, float2 b) {
  return make_float2(a.x - b.x, a.y - b.y);
}

// ---------------------------------------------------------------------------
// Kernel 2: build complex Phi_y/Phi_v from the real Gram blocks, diagonal-
// load, Gauss-Jordan G = Phi_v^{-1} Phi_y, then
// h = (G - lam*I)[:, ref] / (tr(G) - 8*lam); store w = conj(h).
// ---------------------------------------------------------------------------
__global__ void solve_kernel(float* __restrict__ ws,
                             const int* __restrict__ refp) {
  const int tid = blockIdx.x * blockDim.x + threadIdx.x;
  if (tid >= NF) return;
  const int ref = *refp;
  const float* Sy = ws + SY_OFF + (long)tid * 256;
  const float* Sv = ws + SV_OFF + (long)tid * 256;

  const float lam = 0.70710678118654752f;  // (1+1j)/sqrt(2) -> re == im

  float2 A[8][8];  // Phi_v (loaded)
  float2 G[8][8];  // Phi_y, becomes G = Phi_v^{-1} Phi_y
  for (int i = 0; i < 8; ++i) {
    for (int j = 0; j < 8; ++j) {
      // Z = [Yre;Yim]: Re(Phi) = S00 + S11, Im(Phi) = S10 - S01
      float re = Sv[i * 16 + j] + Sv[(i + 8) * 16 + (j + 8)];
      float im = Sv[(i + 8) * 16 + j] - Sv[i * 16 + (j + 8)];
      if (i == j) { re += 0.001f * lam; im += 0.001f * lam; }
      A[i][j] = make_float2(re, im);
      G[i][j] = make_float2(Sy[i * 16 + j] + Sy[(i + 8) * 16 + (j + 8)],
                            Sy[(i + 8) * 16 + j] - Sy[i * 16 + (j + 8)]);
    }
  }

  // Gauss-Jordan without pivoting (A is diagonally-loaded near-HPD).
  for (int k = 0; k < 8; ++k) {
    const float2 p = A[k][k];
    const float d = p.x * p.x + p.y * p.y;
    const float2 ip = make_float2(p.x / d, -p.y / d);
    for (int j = k; j < 8; ++j) A[k][j] = cmul(A[k][j], ip);
    for (int j = 0; j < 8; ++j) G[k][j] = cmul(G[k][j], ip);
    for (int i = 0; i < 8; ++i) {
      if (i == k) continue;
      const float2 fct = A[i][k];
      for (int j = k; j < 8; ++j) A[i][j] = csub(A[i][j], cmul(fct, A[k][j]));
      for (int j = 0; j < 8; ++j) G[i][j] = csub(G[i][j], cmul(fct, G[k][j]));
    }
  }

  // l = trace(G) - 8*lam
  float lr = 0.f, li = 0.f;
  for (int k = 0; k < 8; ++k) { lr += G[k][k].x; li += G[k][k].y; }
  lr -= 8.0f * lam;
  li -= 8.0f * lam;
  const float ld = lr * lr + li * li;

  float2* w = (float2*)(ws + H_OFF) + (long)tid * 8;
  for (int c = 0; c < 8; ++c) {
    float2 num = G[c][ref];
    if (c == ref) { num.x -= lam; num.y -= lam; }
    const float hr = (num.x * lr + num.y * li) / ld;  // h = num / l
    const float hi = (num.y * lr - num.x * li) / ld;
    w[c] = make_float2(hr, -hi);  // store conj(h)
  }
}

// ---------------------------------------------------------------------------
// Kernel 3: beamform X(n,t,f) = sum_c conj(h) * y. Lane == consecutive f so
// every global access is a full 128B line per wave; each thread amortizes its
// 8 complex weights across a 50-frame t-chunk. 41040 threads.
// ---------------------------------------------------------------------------
#define TCH 50
__global__ __launch_bounds__(256) void beamform_kernel(
    const float* __restrict__ mix, const float* __restrict__ ws,
    float* __restrict__ out) {
  const int tid = blockIdx.x * blockDim.x + threadIdx.x;
  if (tid >= NF * (TB / TCH)) return;
  const int f = tid % FB;
  const int rest = tid / FB;
  const int n = rest % NB;
  const int tc = rest / NB;

  const float2* w = (const float2*)(ws + H_OFF) + ((long)n * FB + f) * 8;
  float2 wl[8];
#pragma unroll
  for (int c = 0; c < 8; ++c) wl[c] = w[c];

  const float* pb = mix + (long)n * (2L * CB * TB * FB) + f;
  float* ob = out + (long)n * (2L * TB * FB) + f;

  for (int t = tc * TCH; t < tc * TCH + TCH; ++t) {
    float xr = 0.f, xi = 0.f;
#pragma unroll
    for (int c = 0; c < 8; ++c) {
      const float yr = pb[(long)c * (TB * FB) + (long)t * FB];
      const float yi =
          pb[(long)(CB * TB * FB) + (long)c * (TB * FB) + (long)t * FB];
      xr += wl[c].x * yr - wl[c].y * yi;
      xi += wl[c].x * yi + wl[c].y * yr;
    }
    ob[(long)t * FB] = xr;                    // [n,0,0,t,f]
    ob[(long)(TB * FB) + (long)t * FB] = xi;  // [n,1,0,t,f]
  }
}

// ---------------------------------------------------------------------------
extern "C" void kernel_launch(void* const* d_in, const int* in_sizes, int n_in,
                              void* d_out, int out_size, void* d_ws,
                              size_t ws_size, hipStream_t stream) {
  const float* mix = (const float*)d_in[0];
  // d_in[1] = target (unused in forward)
  const float* noi = (const float*)d_in[2];
  const int* ref = (const int*)d_in[3];
  float* ws = (float*)d_ws;
  float* out = (float*)d_out;

  // 129 blocks * 8 waves = 1032 waves == NB*QN (full EXEC for WMMA)
  cov_wmma_kernel<<<QN, 256, 0, stream>>>(mix, noi, ws);
  solve_kernel<<<(NF + 63) / 64, 64, 0, stream>>>(ws, ref);
  const int total = NF * (TB / TCH);
  beamform_kernel<<<(total + 255) / 256, 256, 0, stream>>>(mix, ws, out);
}